// S6_36180804502182
// MI455X (gfx1250) — compile-verified
//
#include <hip/hip_runtime.h>
#include <math.h>

typedef __attribute__((ext_vector_type(2))) float v2f;
typedef __attribute__((ext_vector_type(8))) float v8f;

#define BSZ 8
#define LEN 4096
#define FD  64
#define NS  16
#define NPOS (BSZ*LEN)  // 32768 (b,l) positions

// workspace layout in floats
#define WS_BM 0                    // (NPOS, NS)   input-dependent B projection
#define WS_D0 (NPOS*NS)            // (NPOS,)      step sizes
#define WS_S  (WS_D0 + NPOS)       // (NPOS,)      suffix sums of d0
#define WS_W  (WS_S + NPOS)        // (NPOS,)      scalar weights w(b,l)
#define WS_C  (WS_W + NPOS)        // (BSZ, NS)    C_last

// ---------------------------------------------------------------------------
// Kernel 1: WMMA f32 projection.  Each wave owns 16 consecutive positions.
//   c  = X(16x64) @ W_B(64x16)          -> Bm tile
//   c2 = X(16x64) @ [W_d | 0 ... 0]     -> column 0 = x . W_d  -> d0 (softplus)
// A 16x4 f32 layout: lanes 0-15 = rows, vgpr{0,1} = K{0,1}; lanes 16-31 = K{2,3}.
// B 4x16 f32 layout mirrors A (rows K striped across lanes / half-waves).
// C/D 16x16 f32: vgpr r = row r (lanes 0-15, N=lane) and row r+8 (lanes 16-31).
// ---------------------------------------------------------------------------
__global__ __launch_bounds__(256) void k_proj(const float* __restrict__ x,
                                              const float* __restrict__ WB,
                                              const float* __restrict__ bB,
                                              const float* __restrict__ Wd,
                                              const float* __restrict__ bd,
                                              float* __restrict__ Bm,
                                              float* __restrict__ d0) {
  const int lane = threadIdx.x & 31;
  const int wv   = threadIdx.x >> 5;          // 8 waves per block
  const int half = lane >> 4;                 // 0: lanes 0-15, 1: lanes 16-31
  const int lr   = lane & 15;                 // row (A) / col (B,C)
  const int tile = blockIdx.x * 8 + wv;       // 2048 tiles of 16 positions
  const int base = tile * 16;

  const float* xrow = x + (size_t)(base + lr) * FD;

  v8f c  = {};
  v8f c2 = {};
  for (int ks = 0; ks < 16; ++ks) {
    const int k = ks * 4 + half * 2;          // this lane's K pair
    v2f a;  a.x = xrow[k];            a.y = xrow[k + 1];
    v2f b;  b.x = WB[k * NS + lr];    b.y = WB[(k + 1) * NS + lr];
    const float w0 = Wd[k], w1 = Wd[k + 1];
    v2f b2; b2.x = (lr == 0) ? w0 : 0.0f;  b2.y = (lr == 0) ? w1 : 0.0f;
    c  = __builtin_amdgcn_wmma_f32_16x16x4_f32(false, a, false, b,
                                               (short)0, c,  false, false);
    c2 = __builtin_amdgcn_wmma_f32_16x16x4_f32(false, a, false, b2,
                                               (short)0, c2, false, false);
  }

  const float bb  = bB[lr];
  const float bdv = bd[0];
#pragma unroll
  for (int r = 0; r < 8; ++r) {
    const int pos = base + r + half * 8;
    Bm[(size_t)pos * NS + lr] = c[r] + bb;
  }
  if (lr == 0) {                               // lanes 0 and 16 hold column 0
#pragma unroll
    for (int r = 0; r < 8; ++r) {
      const int pos = base + r + half * 8;
      const float t = 1.0f /*DELTA*/ + c2[r] + bdv;
      // stable softplus
      d0[pos] = fmaxf(t, 0.0f) + log1pf(__expf(-fabsf(t)));
    }
  }
}

// ---------------------------------------------------------------------------
// Kernel 2: per-batch suffix sums  S(b,l) = sum_{l' > l} d0(b,l').
// One block per batch; 256 threads x 16 contiguous elements each.
// ---------------------------------------------------------------------------
__global__ __launch_bounds__(256) void k_scan(const float* __restrict__ d0,
                                              float* __restrict__ S) {
  const int b = blockIdx.x;
  const int t = threadIdx.x;
  const float* d = d0 + (size_t)b * LEN;
  float*       s = S  + (size_t)b * LEN;

  const int base = t * 16;
  float v[16];
  float lsum = 0.0f;
#pragma unroll
  for (int i = 0; i < 16; ++i) { v[i] = d[base + i]; lsum += v[i]; }

  __shared__ float ps[256];
  ps[t] = lsum;
  __syncthreads();
  // inclusive suffix scan over the 256 partials (Hillis-Steele)
  for (int off = 1; off < 256; off <<= 1) {
    const float add = (t + off < 256) ? ps[t + off] : 0.0f;
    __syncthreads();
    ps[t] += add;
    __syncthreads();
  }
  float run = (t < 255) ? ps[t + 1] : 0.0f;    // exclusive suffix of my chunk
#pragma unroll
  for (int i = 15; i >= 0; --i) {
    s[base + i] = run;                         // strictly-after sum
    run += v[i];
  }
}

// ---------------------------------------------------------------------------
// Kernel 3: C_last(b,n) = x[b, L-1, :] . W_C[:, n] + b_C[n]   (real, 8x16)
// ---------------------------------------------------------------------------
__global__ __launch_bounds__(128) void k_clast(const float* __restrict__ x,
                                               const float* __restrict__ WC,
                                               const float* __restrict__ bC,
                                               float* __restrict__ C) {
  const int t = threadIdx.x;                   // 128 = 8 batches x 16 states
  const int b = t >> 4;
  const int n = t & 15;
  const float* xr = x + ((size_t)b * LEN + (LEN - 1)) * FD;
  float acc = bC[n];
  for (int f = 0; f < FD; ++f) acc += xr[f] * WC[f * NS + n];
  C[b * NS + n] = acc;
}

// ---------------------------------------------------------------------------
// Kernel 4: scalar weights
//   w(b,l) = sum_n C(b,n) * Re( exp(A_n * S) * (dA-1)/dA * d0 * Bm_n )
//   with dA = d0 * A_n (complex), Re(A_n) = -0.5 so exp never overflows.
// ---------------------------------------------------------------------------
__global__ __launch_bounds__(256) void k_weights(const float* __restrict__ d0,
                                                 const float* __restrict__ S,
                                                 const float* __restrict__ Bm,
                                                 const float* __restrict__ C,
                                                 const float* __restrict__ Are,
                                                 const float* __restrict__ Aim,
                                                 float* __restrict__ w) {
  const int p = blockIdx.x * 256 + threadIdx.x;   // (b,l) flat, 32768 total
  const int b = p >> 12;                          // p / LEN
  const float d = d0[p];
  const float s = S[p];
  const float* bm = Bm + (size_t)p * NS;
  float acc = 0.0f;
#pragma unroll
  for (int n = 0; n < NS; ++n) {
    const float ar = Are[n];                      // A tiled over F: row 0
    const float ai = Aim[n];
    const float dar = d * ar, dai = d * ai;
    const float den = dar * dar + dai * dai;      // |dA|^2 > 0
    const float inv = 1.0f / den;
    // coef = (dA - 1)/dA = ((den - dar) + i*dai) / den
    const float cr = (den - dar) * inv;
    const float ci = dai * inv;
    const float db = d * bm[n];                   // dB (real)
    const float Bdr = cr * db, Bdi = ci * db;
    // G = exp(A_n * S) = exp(ar*s) * (cos(ai*s) + i sin(ai*s))
    const float er = __expf(ar * s);
    float sn, cs;
    __sincosf(ai * s, &sn, &cs);
    const float Gr = er * cs, Gi = er * sn;
    acc += C[b * NS + n] * (Gr * Bdr - Gi * Bdi); // Re(G * Bd), C real
  }
  w[p] = acc;
}

// ---------------------------------------------------------------------------
// Kernel 5: y(b,f) = sum_l w(b,l) * x(b,l,f).  One block per batch.
// ---------------------------------------------------------------------------
__global__ __launch_bounds__(256) void k_out(const float* __restrict__ x,
                                             const float* __restrict__ w,
                                             float* __restrict__ y) {
  const int b = blockIdx.x;
  const int t = threadIdx.x;
  const int f = t & 63;
  const int g = t >> 6;                           // 4 L-groups
  const float* xb = x + (size_t)b * LEN * FD;
  const float* wb = w + (size_t)b * LEN;
  float acc = 0.0f;
  for (int l = g; l < LEN; l += 4) acc += wb[l] * xb[(size_t)l * FD + f];
  __shared__ float red[256];
  red[t] = acc;
  __syncthreads();
  if (g == 0) y[b * FD + f] = red[f] + red[64 + f] + red[128 + f] + red[192 + f];
}

// ---------------------------------------------------------------------------
extern "C" void kernel_launch(void* const* d_in, const int* in_sizes, int n_in,
                              void* d_out, int out_size, void* d_ws, size_t ws_size,
                              hipStream_t stream) {
  const float* x   = (const float*)d_in[0];   // (8, 4096, 64)
  const float* Are = (const float*)d_in[1];   // (64, 16) tiled rows
  const float* Aim = (const float*)d_in[2];   // (64, 16)
  const float* WB  = (const float*)d_in[3];   // (64, 16)
  const float* bB  = (const float*)d_in[4];   // (16,)
  const float* WC  = (const float*)d_in[5];   // (64, 16)
  const float* bC  = (const float*)d_in[6];   // (16,)
  const float* Wd  = (const float*)d_in[7];   // (64, 1)
  const float* bd  = (const float*)d_in[8];   // (1,)

  float* ws = (float*)d_ws;
  float* Bm = ws + WS_BM;
  float* d0 = ws + WS_D0;
  float* S  = ws + WS_S;
  float* wv = ws + WS_W;
  float* C  = ws + WS_C;
  float* y  = (float*)d_out;                  // (8, 64)

  k_proj   <<<NPOS / 128, 256, 0, stream>>>(x, WB, bB, Wd, bd, Bm, d0);
  k_scan   <<<BSZ,        256, 0, stream>>>(d0, S);
  k_clast  <<<1,          128, 0, stream>>>(x, WC, bC, C);
  k_weights<<<NPOS / 256, 256, 0, stream>>>(d0, S, Bm, C, Are, Aim, wv);
  k_out    <<<BSZ,        256, 0, stream>>>(x, wv, y);
}